// get_model_57397942943829
// MI455X (gfx1250) — compile-verified
//
#include <hip/hip_runtime.h>
#include <hip/hip_bf16.h>
#include <math.h>

#define B_    8
#define NPTS  2048
#define NEDGE 32768
#define NCLS  50

typedef __attribute__((ext_vector_type(2))) float v2f;
typedef __attribute__((ext_vector_type(8))) float v8f;

__host__ __device__ static inline int a16i(int x) { return (x + 15) & ~15; }

__device__ __forceinline__ v8f wmma_f32x4(v2f a, v2f b, v8f c) {
  // D = A(16x4) * B(4x16) + C, fp32, wave32
  return __builtin_amdgcn_wmma_f32_16x16x4_f32(false, a, false, b, (short)0, c,
                                               false, false);
}

// ---------------------------------------------------------------------------
// transpose (B,3,N) -> (B,N,3)
__global__ void k_transpose_xyz(const float* __restrict__ xyz, float* __restrict__ pts) {
  int g = blockIdx.x * 256 + threadIdx.x;
  int total = B_ * NPTS * 3;
  if (g >= total) return;
  int b = g / (NPTS * 3);
  int rem = g % (NPTS * 3);
  int i = rem / 3, c = rem % 3;
  pts[g] = xyz[(b * 3 + c) * NPTS + i];
}

// ---------------------------------------------------------------------------
// Farthest point sampling: one block per batch. Matches lax.scan semantics:
// idx[i] = far (pre-update), dist = min(dist, d(p, p[far])), far = argmax(dist).
__global__ __launch_bounds__(256) void k_fps(const float* __restrict__ pts, int n,
                                             int npoint, int* __restrict__ fps_idx,
                                             float* __restrict__ new_xyz) {
  __shared__ float dist[NPTS];
  __shared__ float bv[256];
  __shared__ int   bi[256];
  __shared__ int   sidx[512];
  __shared__ int   sfar;
  int b = blockIdx.x, t = threadIdx.x;
  for (int j = t; j < n; j += 256) dist[j] = 1e10f;
  if (t == 0) sfar = 0;
  __syncthreads();
  for (int it = 0; it < npoint; ++it) {
    int far = sfar;
    if (t == 0) sidx[it] = far;
    float fx = pts[(b * n + far) * 3 + 0];
    float fy = pts[(b * n + far) * 3 + 1];
    float fz = pts[(b * n + far) * 3 + 2];
    float lv = -1.0f; int li = 0;
    for (int j = t; j < n; j += 256) {
      float dx = pts[(b * n + j) * 3 + 0] - fx;
      float dy = pts[(b * n + j) * 3 + 1] - fy;
      float dz = pts[(b * n + j) * 3 + 2] - fz;
      float d = dx * dx + dy * dy + dz * dz;
      float nd = fminf(dist[j], d);
      dist[j] = nd;
      if (nd > lv) { lv = nd; li = j; }
    }
    bv[t] = lv; bi[t] = li;
    __syncthreads();
    for (int s = 128; s > 0; s >>= 1) {
      if (t < s) {
        float v2 = bv[t + s]; int i2 = bi[t + s];
        if (v2 > bv[t] || (v2 == bv[t] && i2 < bi[t])) { bv[t] = v2; bi[t] = i2; }
      }
      __syncthreads();
    }
    if (t == 0) sfar = bi[0];
    __syncthreads();
  }
  for (int e = t; e < npoint; e += 256) fps_idx[b * npoint + e] = sidx[e];
  for (int e = t; e < npoint * 3; e += 256) {
    int i = e / 3, c = e % 3;
    new_xyz[(b * npoint + i) * 3 + c] = pts[(b * n + sidx[i]) * 3 + c];
  }
}

// ---------------------------------------------------------------------------
// Ball query: one wave32 per query point; ordered first-k compaction via ballot.
__global__ __launch_bounds__(256) void k_query_ball(const float* __restrict__ xyz,
                                                    const float* __restrict__ nxyz,
                                                    int n, int npoint, float r2, int k,
                                                    int* __restrict__ idxbuf) {
  int gid = blockIdx.x * 8 + (threadIdx.x >> 5);
  if (gid >= B_ * npoint) return;
  int lane = threadIdx.x & 31;
  int b = gid / npoint;
  float cx = nxyz[gid * 3 + 0], cy = nxyz[gid * 3 + 1], cz = nxyz[gid * 3 + 2];
  int cnt = 0, first = -1;
  for (int j0 = 0; j0 < n && cnt < k; j0 += 32) {
    int j = j0 + lane;
    bool in = false;
    if (j < n) {
      float dx = xyz[(b * n + j) * 3 + 0] - cx;
      float dy = xyz[(b * n + j) * 3 + 1] - cy;
      float dz = xyz[(b * n + j) * 3 + 2] - cz;
      in = (dx * dx + dy * dy + dz * dz) <= r2;
    }
    unsigned m = (unsigned)__ballot(in);
    int pos = cnt + __popc(m & ((1u << lane) - 1u));
    if (in && pos < k) idxbuf[gid * k + pos] = j;
    if (first < 0 && m) first = j0 + (__ffs(m) - 1);
    cnt += __popc(m);
  }
  int padv = (first >= 0) ? first : (n - 1);
  for (int r = cnt + lane; r < k; r += 32) idxbuf[gid * k + r] = padv;
}

// ---------------------------------------------------------------------------
// One WMMA MLP layer fully in LDS: actOut = relu((actIn @ W^T + b) * g + bt).
// actIn: k x ldin (zero-padded). W: (dout, kreal) row-major.
// Out-of-range lanes are clamped (not guarded) in the hot loop; epilogue masks.
__device__ __forceinline__ void lds_layer(const float* actIn, int ldin, int kreal,
                                          const float* __restrict__ W,
                                          const float* __restrict__ bs,
                                          const float* __restrict__ gn,
                                          const float* __restrict__ bt, int dout,
                                          float* actOut, int ldo, int kRows) {
  int wv = threadIdx.x >> 5, lane = threadIdx.x & 31;
  int hl = lane >> 4, mr = lane & 15;
  int tilesN = (dout + 15) >> 4;
  int tilesM = kRows >> 4;
  int kmain = kreal & ~3;
  for (int tile = wv; tile < tilesM * tilesN; tile += 8) {
    int tm = tile / tilesN, tn = tile % tilesN;
    int row = tm * 16 + mr, col = tn * 16 + mr;
    int colc = (col < dout) ? col : (dout - 1);  // clamp; epilogue masks pad cols
    v8f acc = {0.f, 0.f, 0.f, 0.f, 0.f, 0.f, 0.f, 0.f};
    const float* arow = actIn + row * ldin + hl * 2;
    const float* wrow = W + (long)colc * kreal + hl * 2;
#pragma unroll 4
    for (int kk = 0; kk < kmain; kk += 4) {
      v2f A; A.x = arow[kk]; A.y = arow[kk + 1];
      v2f Bv; Bv.x = wrow[kk]; Bv.y = wrow[kk + 1];
      acc = wmma_f32x4(A, Bv, acc);
    }
    if (kmain < kreal) {  // single guarded tail chunk
      int ka = kmain + hl * 2;
      v2f A; A.x = arow[kmain]; A.y = arow[kmain + 1];  // LDS pad is zero
      v2f Bv;
      Bv.x = (ka < kreal) ? wrow[kmain] : 0.f;
      Bv.y = (ka + 1 < kreal) ? wrow[kmain + 1] : 0.f;
      acc = wmma_f32x4(A, Bv, acc);
    }
    float bb = bs[colc], gg = gn[colc], tt = bt[colc];
    for (int v = 0; v < 8; ++v) {
      int m = tm * 16 + v + hl * 8;
      int nn = tn * 16 + mr;
      float val = (nn < dout) ? fmaxf(0.f, (acc[v] + bb) * gg + tt) : 0.f;
      actOut[m * ldo + nn] = val;
    }
  }
}

// Fused gather + 3-layer MLP + max-pool per group, all resident in LDS.
__global__ __launch_bounds__(256) void k_group_mlp_max(
    const float* __restrict__ points, const float* __restrict__ xyz,
    const float* __restrict__ nxyz, const int* __restrict__ idxbuf,
    const float* W0, const float* b0, const float* g0, const float* t0, int d0,
    const float* W1, const float* b1, const float* g1, const float* t1, int d1,
    const float* W2, const float* b2, const float* g2, const float* t2, int d2,
    float* __restrict__ out, int ldOut, int colOff, int k, int npoint, int nsrc,
    int cpts, int szA) {
  extern __shared__ float lds[];
  float* regA = lds;
  float* regB = lds + szA;
  int gid = blockIdx.x;
  int b = gid / npoint;
  int cin = cpts + 3;
  int cinP = a16i(cin), d0P = a16i(d0), d1P = a16i(d1), d2P = a16i(d2);
  // gather grouped features into LDS: [points(idx) | xyz(idx)-center | 0pad]
  for (int e = threadIdx.x; e < k * cinP; e += 256) {
    int r = e / cinP, c = e % cinP;
    int id = idxbuf[gid * k + r];
    if (id >= nsrc) id = nsrc - 1;
    float v = 0.f;
    if (c < cpts) v = points[((long)(b * nsrc + id)) * cpts + c];
    else if (c < cin) {
      int cc = c - cpts;
      v = xyz[(b * nsrc + id) * 3 + cc] - nxyz[gid * 3 + cc];
    }
    regB[e] = v;
  }
  __syncthreads();
  lds_layer(regB, cinP, cin, W0, b0, g0, t0, d0, regA, d0P, k);
  __syncthreads();
  lds_layer(regA, d0P, d0, W1, b1, g1, t1, d1, regB, d1P, k);
  __syncthreads();
  lds_layer(regB, d1P, d1, W2, b2, g2, t2, d2, regA, d2P, k);
  __syncthreads();
  for (int c = threadIdx.x; c < d2; c += 256) {
    float m = regA[c];
    for (int r = 1; r < k; ++r) m = fmaxf(m, regA[r * d2P + c]);
    out[(long)gid * ldOut + colOff + c] = m;
  }
}

// ---------------------------------------------------------------------------
// Generic WMMA fp32 GEMM: Y = epi(X @ op(W)). WT: W is (dout,din) -> X@W^T.
// EPI: 0 none, 1 +bias, 2 relu((x+b)*g+bt). One wave per 16x16 tile.
// Main K loop is branch-free: OOR lanes load from clamped rows, epilogue masks.
template <int EPI, bool WT>
__global__ __launch_bounds__(256) void k_gemm(const float* __restrict__ X,
                                              const float* __restrict__ W,
                                              const float* __restrict__ bs,
                                              const float* __restrict__ gn,
                                              const float* __restrict__ bt,
                                              float* __restrict__ Y, int rows, int din,
                                              int dout, int ldx, int ldy) {
  int tilesN = (dout + 15) >> 4;
  int tilesM = (rows + 15) >> 4;
  int tile = blockIdx.x * 8 + (threadIdx.x >> 5);
  if (tile >= tilesM * tilesN) return;
  int lane = threadIdx.x & 31, hl = lane >> 4, mr = lane & 15;
  int tm = tile / tilesN, tn = tile % tilesN;
  int row = tm * 16 + mr, col = tn * 16 + mr;
  int rowc = (row < rows) ? row : (rows - 1);
  int colc = (col < dout) ? col : (dout - 1);
  v8f acc = {0.f, 0.f, 0.f, 0.f, 0.f, 0.f, 0.f, 0.f};
  const float* xr = X + (long)rowc * ldx + hl * 2;
  const float* wr = WT ? (W + (long)colc * din + hl * 2) : (W + colc + (long)hl * 2 * dout);
  int kmain = din & ~3;
#pragma unroll 4
  for (int kk = 0; kk < kmain; kk += 4) {
    v2f A; A.x = xr[kk]; A.y = xr[kk + 1];
    v2f Bv;
    if (WT) { Bv.x = wr[kk]; Bv.y = wr[kk + 1]; }
    else    { Bv.x = wr[(long)kk * dout]; Bv.y = wr[(long)(kk + 1) * dout]; }
    acc = wmma_f32x4(A, Bv, acc);
  }
  if (kmain < din) {  // single guarded tail chunk
    int ka = kmain + hl * 2;
    v2f A; v2f Bv;
    A.x = (ka < din) ? xr[kmain] : 0.f;
    A.y = (ka + 1 < din) ? xr[kmain + 1] : 0.f;
    if (WT) {
      Bv.x = (ka < din) ? wr[kmain] : 0.f;
      Bv.y = (ka + 1 < din) ? wr[kmain + 1] : 0.f;
    } else {
      Bv.x = (ka < din) ? wr[(long)kmain * dout] : 0.f;
      Bv.y = (ka + 1 < din) ? wr[(long)(kmain + 1) * dout] : 0.f;
    }
    acc = wmma_f32x4(A, Bv, acc);
  }
  float bb = 0.f, gg = 1.f, tt = 0.f;
  if (EPI >= 1) bb = bs[colc];
  if (EPI == 2) { gg = gn[colc]; tt = bt[colc]; }
  for (int v = 0; v < 8; ++v) {
    int m = tm * 16 + v + hl * 8;
    int nn = tn * 16 + mr;
    if (m < rows && nn < dout) {
      float val = acc[v];
      if (EPI == 1) val += bb;
      else if (EPI == 2) val = fmaxf(0.f, (val + bb) * gg + tt);
      Y[(long)m * ldy + nn] = val;
    }
  }
}

// ---------------------------------------------------------------------------
__global__ void k_colmax(const float* __restrict__ in, float* __restrict__ out, int k,
                         int d) {
  int g = blockIdx.x * 256 + threadIdx.x;
  if (g >= B_ * d) return;
  int b = g / d, c = g % d;
  const float* p = in + (long)b * k * d + c;
  float m = p[0];
  for (int r = 1; r < k; ++r) m = fmaxf(m, p[(long)r * d]);
  out[g] = m;
}

__global__ void k_build_sa3x(const float* __restrict__ nx2, const float* __restrict__ l2p,
                             float* __restrict__ X) {
  int g = blockIdx.x * 256 + threadIdx.x;
  int total = B_ * 128 * 515;
  if (g >= total) return;
  int row = g / 515, c = g % 515;
  X[g] = (c < 3) ? nx2[row * 3 + c] : l2p[(long)row * 512 + (c - 3)];
}

__global__ void k_build_fp3x(const float* __restrict__ l2p, const float* __restrict__ l3,
                             float* __restrict__ X) {
  int g = blockIdx.x * 256 + threadIdx.x;
  int total = B_ * 128 * 1536;
  if (g >= total) return;
  int row = g / 1536, c = g % 1536;
  X[g] = (c < 512) ? l2p[(long)row * 512 + c] : l3[(row / 128) * 1024 + (c - 512)];
}

// 3-NN + inverse-distance weights (matches top_k(-d,3) ordering)
__global__ void k_knn3(const float* __restrict__ q, const float* __restrict__ ref, int nq,
                       int nref, float* __restrict__ w3, int* __restrict__ id3) {
  int g = blockIdx.x * 256 + threadIdx.x;
  if (g >= B_ * nq) return;
  int b = g / nq;
  float qx = q[g * 3], qy = q[g * 3 + 1], qz = q[g * 3 + 2];
  float d0 = 3.4e38f, d1 = 3.4e38f, d2 = 3.4e38f;
  int i0 = 0, i1 = 0, i2 = 0;
  const float* rb = ref + (long)b * nref * 3;
  for (int j = 0; j < nref; ++j) {
    float dx = qx - rb[j * 3], dy = qy - rb[j * 3 + 1], dz = qz - rb[j * 3 + 2];
    float d = dx * dx + dy * dy + dz * dz;
    if (d < d0)      { d2 = d1; i2 = i1; d1 = d0; i1 = i0; d0 = d; i0 = j; }
    else if (d < d1) { d2 = d1; i2 = i1; d1 = d; i1 = j; }
    else if (d < d2) { d2 = d; i2 = j; }
  }
  float r0 = 1.f / (d0 + 1e-8f), r1 = 1.f / (d1 + 1e-8f), r2 = 1.f / (d2 + 1e-8f);
  float s = r0 + r1 + r2;
  w3[g * 3] = r0 / s; w3[g * 3 + 1] = r1 / s; w3[g * 3 + 2] = r2 / s;
  id3[g * 3] = i0; id3[g * 3 + 1] = i1; id3[g * 3 + 2] = i2;
}

__global__ void k_build_fp2x(const float* __restrict__ l1p, const float* __restrict__ l2pf,
                             const float* __restrict__ w3, const int* __restrict__ id3,
                             float* __restrict__ X) {
  long g = (long)blockIdx.x * 256 + threadIdx.x;
  long total = (long)B_ * 512 * 576;
  if (g >= total) return;
  int row = (int)(g / 576), c = (int)(g % 576);
  float v;
  if (c < 320) v = l1p[(long)row * 320 + c];
  else {
    int b = row / 512, cc = c - 320;
    v = 0.f;
    for (int j = 0; j < 3; ++j)
      v += w3[row * 3 + j] * l2pf[((long)(b * 128 + id3[row * 3 + j])) * 256 + cc];
  }
  X[g] = v;
}

__global__ void k_build_fp1x(const float* __restrict__ cls, const float* __restrict__ pts,
                             const float* __restrict__ l1f, const float* __restrict__ w3,
                             const int* __restrict__ id3, float* __restrict__ X) {
  long g = (long)blockIdx.x * 256 + threadIdx.x;
  long total = (long)B_ * NPTS * 150;
  if (g >= total) return;
  int row = (int)(g / 150), c = (int)(g % 150);
  int b = row / NPTS;
  float v;
  if (c < 16) v = cls[b * 16 + c];
  else if (c < 19) v = pts[(long)row * 3 + (c - 16)];
  else if (c < 22) v = pts[(long)row * 3 + (c - 19)];
  else {
    int cc = c - 22;
    v = 0.f;
    for (int j = 0; j < 3; ++j)
      v += w3[row * 3 + j] * l1f[((long)(b * 512 + id3[row * 3 + j])) * 128 + cc];
  }
  X[g] = v;
}

// ---------------------------------------------------------------------------
__global__ void k_fill(float* __restrict__ p, float v, long n) {
  long g = (long)blockIdx.x * 256 + threadIdx.x;
  if (g < n) p[g] = v;
}
__global__ void k_deg_scatter(const int* __restrict__ pg, float* __restrict__ deg) {
  long g = (long)blockIdx.x * 256 + threadIdx.x;
  if (g >= (long)B_ * NEDGE) return;
  int b = (int)(g / NEDGE), e = (int)(g % NEDGE);
  int dst = pg[((long)b * 2 + 1) * NEDGE + e];
  atomicAdd(&deg[b * NPTS + dst], 1.0f);
}
__global__ void k_rsqrt(const float* __restrict__ in, float* __restrict__ out, long n) {
  long g = (long)blockIdx.x * 256 + threadIdx.x;
  if (g < n) out[g] = rsqrtf(in[g]);
}
__global__ void k_gcn_init(const float* __restrict__ h, const float* __restrict__ dinv,
                           const float* __restrict__ bias, float* __restrict__ out) {
  long g = (long)blockIdx.x * 256 + threadIdx.x;
  long total = (long)B_ * NPTS * 128;
  if (g >= total) return;
  int i = (int)(g / 128), c = (int)(g % 128);
  float dv = dinv[i];
  out[g] = bias[c] + h[g] * dv * dv;  // self-loop contribution + bias
}
__global__ __launch_bounds__(128) void k_gcn_edges(const int* __restrict__ pg,
                                                   const float* __restrict__ h,
                                                   const float* __restrict__ dinv,
                                                   float* __restrict__ out) {
  int be = blockIdx.x;
  int b = be / NEDGE, e = be % NEDGE;
  int src = pg[((long)b * 2 + 0) * NEDGE + e];
  int dst = pg[((long)b * 2 + 1) * NEDGE + e];
  float norm = dinv[b * NPTS + src] * dinv[b * NPTS + dst];
  int c = threadIdx.x;
  atomicAdd(&out[((long)(b * NPTS + dst)) * 128 + c],
            h[((long)(b * NPTS + src)) * 128 + c] * norm);
}
__global__ void k_relu(float* __restrict__ p, long n) {
  long g = (long)blockIdx.x * 256 + threadIdx.x;
  if (g < n) p[g] = fmaxf(p[g], 0.f);
}
__global__ void k_logsoftmax(const float* __restrict__ logits, float* __restrict__ out) {
  int row = blockIdx.x * 256 + threadIdx.x;
  if (row >= B_ * NPTS) return;
  const float* xr = logits + (long)row * NCLS;
  float mx = xr[0];
  for (int c = 1; c < NCLS; ++c) mx = fmaxf(mx, xr[c]);
  float s = 0.f;
  for (int c = 0; c < NCLS; ++c) s += expf(xr[c] - mx);
  float ls = logf(s);
  for (int c = 0; c < NCLS; ++c) out[(long)row * NCLS + c] = xr[c] - mx - ls;
}
__global__ void k_copy(const float* __restrict__ in, float* __restrict__ out, long n) {
  long g = (long)blockIdx.x * 256 + threadIdx.x;
  if (g < n) out[g] = in[g];
}

// ---------------------------------------------------------------------------
struct Layer { const float *W, *b, *bt, *g; };

extern "C" void kernel_launch(void* const* d_in, const int* in_sizes, int n_in,
                              void* d_out, int out_size, void* d_ws, size_t ws_size,
                              hipStream_t stream) {
  (void)in_sizes; (void)n_in; (void)out_size; (void)ws_size;
  const float* xyz_in = (const float*)d_in[0];
  const float* cls = (const float*)d_in[1];
  const int* pg = (const int*)d_in[2];

  // params: pytree leaf order (dict keys sorted): fp1,fp2,fp3,gcn,head,sa1,sa2,sa3
  // layer dict order: W, b, bt, g
  int ci = 3;
  auto nf = [&]() { return (const float*)d_in[ci++]; };
  auto rl = [&]() { Layer L; L.W = nf(); L.b = nf(); L.bt = nf(); L.g = nf(); return L; };
  Layer fp1L[2]; for (int i = 0; i < 2; ++i) fp1L[i] = rl();
  Layer fp2L[2]; for (int i = 0; i < 2; ++i) fp2L[i] = rl();
  Layer fp3L[2]; for (int i = 0; i < 2; ++i) fp3L[i] = rl();
  const float* gW1 = nf(); const float* gW2 = nf();
  const float* gb1 = nf(); const float* gb2 = nf();
  const float* hW1 = nf(); const float* hW2 = nf();
  const float* hb1 = nf(); const float* hb2 = nf();
  const float* hbt1 = nf(); const float* hg1 = nf();
  Layer sa1L[3][3];
  for (int m = 0; m < 3; ++m) for (int l = 0; l < 3; ++l) sa1L[m][l] = rl();
  Layer sa2L[2][3];
  for (int m = 0; m < 2; ++m) for (int l = 0; l < 3; ++l) sa2L[m][l] = rl();
  Layer sa3L[3]; for (int l = 0; l < 3; ++l) sa3L[l] = rl();

  // workspace bump allocator
  char* wp = (char*)d_ws;
  auto alloc = [&](size_t bytes) { void* p = wp; wp += (bytes + 255) & ~(size_t)255; return p; };
  float* pts = (float*)alloc((size_t)B_ * NPTS * 3 * 4);
  int* fidx1 = (int*)alloc((size_t)B_ * 512 * 4);
  float* nx1 = (float*)alloc((size_t)B_ * 512 * 3 * 4);
  int* idxqb = (int*)alloc((size_t)B_ * 512 * 128 * 4);
  float* l1p = (float*)alloc((size_t)B_ * 512 * 320 * 4);
  int* fidx2 = (int*)alloc((size_t)B_ * 128 * 4);
  float* nx2 = (float*)alloc((size_t)B_ * 128 * 3 * 4);
  float* l2p = (float*)alloc((size_t)B_ * 128 * 512 * 4);
  float* sa3X = (float*)alloc((size_t)B_ * 128 * 515 * 4);
  float* sa3a = (float*)alloc((size_t)B_ * 128 * 256 * 4);
  float* sa3b = (float*)alloc((size_t)B_ * 128 * 512 * 4);
  float* sa3c = (float*)alloc((size_t)B_ * 128 * 1024 * 4);
  float* l3 = (float*)alloc((size_t)B_ * 1024 * 4);
  float* fp3X = (float*)alloc((size_t)B_ * 128 * 1536 * 4);
  float* fp3a = (float*)alloc((size_t)B_ * 128 * 256 * 4);
  float* l2pf = (float*)alloc((size_t)B_ * 128 * 256 * 4);
  float* w3a = (float*)alloc((size_t)B_ * 512 * 3 * 4);
  int* id3a = (int*)alloc((size_t)B_ * 512 * 3 * 4);
  float* fp2X = (float*)alloc((size_t)B_ * 512 * 576 * 4);
  float* fp2a = (float*)alloc((size_t)B_ * 512 * 256 * 4);
  float* l1pf = (float*)alloc((size_t)B_ * 512 * 128 * 4);
  float* w3b = (float*)alloc((size_t)B_ * NPTS * 3 * 4);
  int* id3b = (int*)alloc((size_t)B_ * NPTS * 3 * 4);
  float* fp1X = (float*)alloc((size_t)B_ * NPTS * 150 * 4);
  float* fp1a = (float*)alloc((size_t)B_ * NPTS * 128 * 4);
  float* l0f = (float*)alloc((size_t)B_ * NPTS * 128 * 4);
  float* deg = (float*)alloc((size_t)B_ * NPTS * 4);
  float* dinv = (float*)alloc((size_t)B_ * NPTS * 4);
  float* hbuf = (float*)alloc((size_t)B_ * NPTS * 128 * 4);
  float* x1 = (float*)alloc((size_t)B_ * NPTS * 128 * 4);
  float* x2 = (float*)alloc((size_t)B_ * NPTS * 128 * 4);
  float* headf = (float*)alloc((size_t)B_ * NPTS * 128 * 4);
  float* logits = (float*)alloc((size_t)B_ * NPTS * NCLS * 4);

  float* out_ls = (float*)d_out;                      // (B,N,50) log-softmax
  float* out_l3 = out_ls + (size_t)B_ * NPTS * NCLS;  // (B,1024,1) == l3 flat

  auto g1 = [](long n) { return dim3((unsigned)((n + 255) / 256)); };

  auto gemm_bn = [&](const float* X, const Layer& L, float* Y, int rows, int din,
                     int dout, int ldx, int ldy) {
    int tiles = ((rows + 15) / 16) * ((dout + 15) / 16);
    k_gemm<2, true><<<dim3((tiles + 7) / 8), dim3(256), 0, stream>>>(
        X, L.W, L.b, L.g, L.bt, Y, rows, din, dout, ldx, ldy);
  };

  int maxShmem = (128 * a16i(256) + 128 * a16i(323)) * 4;  // worst case sa2 branch 2
  (void)hipFuncSetAttribute((const void*)k_group_mlp_max,
                            hipFuncAttributeMaxDynamicSharedMemorySize, maxShmem);

  auto sa_branch = [&](const float* points, int cpts, int nsrc, const float* sxyz,
                       const float* nxyz, float radius, int k, const Layer* L, int d0,
                       int d1, int d2, float* out, int ldOut, int colOff, int npoint) {
    int blocks_qb = (B_ * npoint + 7) / 8;
    k_query_ball<<<dim3(blocks_qb), dim3(256), 0, stream>>>(sxyz, nxyz, nsrc, npoint,
                                                            radius * radius, k, idxqb);
    int cin = cpts + 3;
    int szA = k * (a16i(d0) > a16i(d2) ? a16i(d0) : a16i(d2));
    int szB = k * (a16i(cin) > a16i(d1) ? a16i(cin) : a16i(d1));
    size_t shmem = (size_t)(szA + szB) * 4;
    k_group_mlp_max<<<dim3(B_ * npoint), dim3(256), shmem, stream>>>(
        points, sxyz, nxyz, idxqb, L[0].W, L[0].b, L[0].g, L[0].bt, d0, L[1].W, L[1].b,
        L[1].g, L[1].bt, d1, L[2].W, L[2].b, L[2].g, L[2].bt, d2, out, ldOut, colOff, k,
        npoint, nsrc, cpts, szA);
  };

  // ---- l0 ----
  k_transpose_xyz<<<g1((long)B_ * NPTS * 3), dim3(256), 0, stream>>>(xyz_in, pts);
  // ---- sa1 (MSG) ----
  k_fps<<<dim3(B_), dim3(256), 0, stream>>>(pts, NPTS, 512, fidx1, nx1);
  sa_branch(pts, 3, NPTS, pts, nx1, 0.1f, 32, sa1L[0], 32, 32, 64, l1p, 320, 0, 512);
  sa_branch(pts, 3, NPTS, pts, nx1, 0.2f, 64, sa1L[1], 64, 64, 128, l1p, 320, 64, 512);
  sa_branch(pts, 3, NPTS, pts, nx1, 0.4f, 128, sa1L[2], 64, 96, 128, l1p, 320, 192, 512);
  // ---- sa2 (MSG) ----
  k_fps<<<dim3(B_), dim3(256), 0, stream>>>(nx1, 512, 128, fidx2, nx2);
  sa_branch(l1p, 320, 512, nx1, nx2, 0.4f, 64, sa2L[0], 128, 128, 256, l2p, 512, 0, 128);
  sa_branch(l1p, 320, 512, nx1, nx2, 0.8f, 128, sa2L[1], 128, 196, 256, l2p, 512, 256, 128);
  // ---- sa3 (global) ----
  k_build_sa3x<<<g1((long)B_ * 128 * 515), dim3(256), 0, stream>>>(nx2, l2p, sa3X);
  gemm_bn(sa3X, sa3L[0], sa3a, B_ * 128, 515, 256, 515, 256);
  gemm_bn(sa3a, sa3L[1], sa3b, B_ * 128, 256, 512, 256, 512);
  gemm_bn(sa3b, sa3L[2], sa3c, B_ * 128, 512, 1024, 512, 1024);
  k_colmax<<<g1((long)B_ * 1024), dim3(256), 0, stream>>>(sa3c, l3, 128, 1024);
  k_copy<<<g1((long)B_ * 1024), dim3(256), 0, stream>>>(l3, out_l3, (long)B_ * 1024);
  // ---- fp3 ----
  k_build_fp3x<<<g1((long)B_ * 128 * 1536), dim3(256), 0, stream>>>(l2p, l3, fp3X);
  gemm_bn(fp3X, fp3L[0], fp3a, B_ * 128, 1536, 256, 1536, 256);
  gemm_bn(fp3a, fp3L[1], l2pf, B_ * 128, 256, 256, 256, 256);
  // ---- fp2 ----
  k_knn3<<<g1((long)B_ * 512), dim3(256), 0, stream>>>(nx1, nx2, 512, 128, w3a, id3a);
  k_build_fp2x<<<g1((long)B_ * 512 * 576), dim3(256), 0, stream>>>(l1p, l2pf, w3a, id3a, fp2X);
  gemm_bn(fp2X, fp2L[0], fp2a, B_ * 512, 576, 256, 576, 256);
  gemm_bn(fp2a, fp2L[1], l1pf, B_ * 512, 256, 128, 256, 128);
  // ---- fp1 ----
  k_knn3<<<g1((long)B_ * NPTS), dim3(256), 0, stream>>>(pts, nx1, NPTS, 512, w3b, id3b);
  k_build_fp1x<<<g1((long)B_ * NPTS * 150), dim3(256), 0, stream>>>(cls, pts, l1pf, w3b,
                                                                    id3b, fp1X);
  gemm_bn(fp1X, fp1L[0], fp1a, B_ * NPTS, 150, 128, 150, 128);
  gemm_bn(fp1a, fp1L[1], l0f, B_ * NPTS, 128, 128, 128, 128);
  // ---- GCN degree/norm (shared by both convs) ----
  k_fill<<<g1((long)B_ * NPTS), dim3(256), 0, stream>>>(deg, 1.0f, (long)B_ * NPTS);
  k_deg_scatter<<<g1((long)B_ * NEDGE), dim3(256), 0, stream>>>(pg, deg);
  k_rsqrt<<<g1((long)B_ * NPTS), dim3(256), 0, stream>>>(deg, dinv, (long)B_ * NPTS);
  // ---- GCN conv1 + relu ----
  {
    int tiles = ((B_ * NPTS + 15) / 16) * (128 / 16);
    k_gemm<0, false><<<dim3((tiles + 7) / 8), dim3(256), 0, stream>>>(
        l0f, gW1, nullptr, nullptr, nullptr, hbuf, B_ * NPTS, 128, 128, 128, 128);
  }
  k_gcn_init<<<g1((long)B_ * NPTS * 128), dim3(256), 0, stream>>>(hbuf, dinv, gb1, x1);
  k_gcn_edges<<<dim3(B_ * NEDGE), dim3(128), 0, stream>>>(pg, hbuf, dinv, x1);
  k_relu<<<g1((long)B_ * NPTS * 128), dim3(256), 0, stream>>>(x1, (long)B_ * NPTS * 128);
  // ---- GCN conv2 ----
  {
    int tiles = ((B_ * NPTS + 15) / 16) * (128 / 16);
    k_gemm<0, false><<<dim3((tiles + 7) / 8), dim3(256), 0, stream>>>(
        x1, gW2, nullptr, nullptr, nullptr, hbuf, B_ * NPTS, 128, 128, 128, 128);
  }
  k_gcn_init<<<g1((long)B_ * NPTS * 128), dim3(256), 0, stream>>>(hbuf, dinv, gb2, x2);
  k_gcn_edges<<<dim3(B_ * NEDGE), dim3(128), 0, stream>>>(pg, hbuf, dinv, x2);
  // ---- head ----
  {
    Layer hL; hL.W = hW1; hL.b = hb1; hL.g = hg1; hL.bt = hbt1;
    gemm_bn(x2, hL, headf, B_ * NPTS, 128, 128, 128, 128);
    int tiles = ((B_ * NPTS + 15) / 16) * ((NCLS + 15) / 16);
    k_gemm<1, true><<<dim3((tiles + 7) / 8), dim3(256), 0, stream>>>(
        headf, hW2, hb2, nullptr, nullptr, logits, B_ * NPTS, 128, NCLS, 128, NCLS);
  }
  k_logsoftmax<<<g1((long)B_ * NPTS), dim3(256), 0, stream>>>(logits, out_ls);
}